// MambaDepthCell_17721035064031
// MI455X (gfx1250) — compile-verified
//
#include <hip/hip_runtime.h>
#include <hip/hip_bf16.h>
#include <math.h>

// Problem constants (from the reference)
constexpr int Tt = 1024;         // B_*N_ tokens
constexpr int Hh = 768;          // hidden
constexpr int Dd = 1536;         // E*H
constexpr int Ss = 16;           // ssm states
constexpr int Kk = 4;            // conv width
constexpr int Rr = 48;           // ceil(H/16)
constexpr int Ll = 2;            // layers
constexpr int PROJN = 2 * Dd;    // 3072
constexpr int SPN = Rr + 2 * Ss; // 80

typedef __attribute__((ext_vector_type(2))) float v2f;
typedef __attribute__((ext_vector_type(4))) float v4f;
typedef __attribute__((ext_vector_type(8))) float v8f;

__device__ __forceinline__ float silu_f(float x) {
  return x / (1.f + __expf(-x));
}

// ---------------------------------------------------------------------------
// RMS norm: one block per token row, 256 threads (8 waves, wave32)
// ---------------------------------------------------------------------------
__global__ void rmsnorm_kernel(const float* __restrict__ x,
                               const float* __restrict__ w,
                               float* __restrict__ out, int Hn) {
  const int t = blockIdx.x;
  const float* xr = x + (size_t)t * Hn;
  float s = 0.f;
  for (int i = threadIdx.x; i < Hn; i += blockDim.x) {
    float v = xr[i];
    s += v * v;
  }
#pragma unroll
  for (int off = 16; off > 0; off >>= 1) s += __shfl_xor(s, off, 32);
  __shared__ float red[8];
  __shared__ float scale;
  if ((threadIdx.x & 31) == 0) red[threadIdx.x >> 5] = s;
  __syncthreads();
  if (threadIdx.x == 0) {
    float tot = 0.f;
    for (int i = 0; i < 8; ++i) tot += red[i];
    scale = rsqrtf(tot / (float)Hn + 1e-5f);
  }
  __syncthreads();
  float* orow = out + (size_t)t * Hn;
  for (int i = threadIdx.x; i < Hn; i += blockDim.x)
    orow[i] = xr[i] * scale * w[i];
}

// ---------------------------------------------------------------------------
// WMMA fp32 GEMM: C[M,N] = A[M,K(lda)] * W[N,K]^T.
// Each wave computes a 16 x (16*NB) strip: NB accumulators share one A
// fragment per k-step (5 b64 loads : NB=4 WMMAs instead of 8 : 4).
// V_WMMA_F32_16X16X4_F32 fragment layout (ISA 7.12.2):
//   A 16x4:  lanes 0-15 row=lane hold K={k,k+1}; lanes 16-31 hold K={k+2,k+3}
//   B 4x16:  lanes 0-15 col=lane hold K={k,k+1}; lanes 16-31 hold K={k+2,k+3}
//   C/D:     lane = col (mod 16), VGPR r = row r (lanes<16) / row 8+r
// MODE: 0 = store, 1 = softplus(acc + bias[col]), 2 = C += acc (residual)
// ---------------------------------------------------------------------------
template <int MODE, int NB>
__global__ void wmma_gemm_nt(const float* __restrict__ A,
                             const float* __restrict__ W,
                             float* __restrict__ C,
                             const float* __restrict__ bias,
                             int M, int N, int Kt, int lda) {
  const int lane = threadIdx.x & 31;
  const int wave = threadIdx.x >> 5;
  const int nStrips = N / (16 * NB);
  const int strip = blockIdx.x * 8 + wave;
  if (strip >= (M >> 4) * nStrips) return;  // wave-uniform: EXEC stays all-1s
  const int m0 = (strip / nStrips) << 4;
  const int n0 = (strip % nStrips) * (16 * NB);
  const int koff = (lane >= 16) ? 2 : 0;

  const float* arow = A + (size_t)(m0 + (lane & 15)) * lda + koff;
  const float* wrow = W + (size_t)(n0 + (lane & 15)) * Kt + koff;
  const size_t wstep = (size_t)16 * Kt;  // next 16-column tile of W

  v8f acc[NB];
#pragma unroll
  for (int nb = 0; nb < NB; ++nb)
    acc[nb] = (v8f){0.f, 0.f, 0.f, 0.f, 0.f, 0.f, 0.f, 0.f};

#pragma unroll 2
  for (int k = 0; k < Kt; k += 4) {
    v2f a = *(const v2f*)(arow + k);
#pragma unroll
    for (int nb = 0; nb < NB; ++nb) {
      v2f b = *(const v2f*)(wrow + (size_t)nb * wstep + k);
      acc[nb] = __builtin_amdgcn_wmma_f32_16x16x4_f32(
          /*neg_a=*/false, a, /*neg_b=*/false, b,
          /*c_mod=*/(short)0, acc[nb], /*reuse_a=*/false, /*reuse_b=*/false);
    }
  }

  const int rbase = m0 + ((lane >= 16) ? 8 : 0);
#pragma unroll
  for (int nb = 0; nb < NB; ++nb) {
    const int col = n0 + nb * 16 + (lane & 15);
#pragma unroll
    for (int r = 0; r < 8; ++r) {
      float v = acc[nb][r];
      float* p = C + (size_t)(rbase + r) * N + col;
      if (MODE == 0) {
        *p = v;
      } else if (MODE == 1) {
        float xx = v + bias[col];
        *p = (xx > 20.f) ? xx : log1pf(__expf(xx));  // softplus
      } else {
        *p += v;  // residual accumulate
      }
    }
  }
}

// ---------------------------------------------------------------------------
// Depthwise conv (K=4) + SiLU; emits new conv state. One thread per (t,d).
// State streams are read-once/write-once and bigger than L2 in aggregate:
// use non-temporal 128-bit accesses so they don't evict the GEMM working set.
// ---------------------------------------------------------------------------
__global__ void conv_kernel(const float* __restrict__ proj,
                            const float* __restrict__ cs_in,
                            const float* __restrict__ conv_w,
                            const float* __restrict__ conv_b,
                            float* __restrict__ hs_conv,
                            float* __restrict__ cs_out) {
  const int idx = blockIdx.x * blockDim.x + threadIdx.x;
  if (idx >= Tt * Dd) return;
  const int t = idx / Dd;
  const int d = idx - t * Dd;
  v4f cin = __builtin_nontemporal_load((const v4f*)cs_in + idx);
  float hsv = proj[(size_t)t * PROJN + d];
  v4f wv = ((const v4f*)conv_w)[d];
  float s = cin.y * wv.x + cin.z * wv.y + cin.w * wv.z + hsv * wv.w + conv_b[d];
  hs_conv[idx] = silu_f(s);
  v4f co = {cin.y, cin.z, cin.w, hsv};
  __builtin_nontemporal_store(co, (v4f*)cs_out + idx);
}

// ---------------------------------------------------------------------------
// SSM state update (the bandwidth-critical kernel): per (t,d) stream 16 states
// in + 16 out as 4x v4f (non-temporal: 400 MB total, > 192 MB L2);
// fuse y = <ss,C> + D*hs and the gate SiLU.
// ---------------------------------------------------------------------------
__global__ void ssm_kernel(const float* __restrict__ hs_conv,
                           const float* __restrict__ dt,
                           const float* __restrict__ sp,
                           const float* __restrict__ proj,
                           const float* __restrict__ ssm_in,
                           const float* __restrict__ A_log,
                           const float* __restrict__ D_param,
                           float* __restrict__ yg,
                           float* __restrict__ ss_out) {
  const int idx = blockIdx.x * blockDim.x + threadIdx.x;
  if (idx >= Tt * Dd) return;
  const int t = idx / Dd;
  const int d = idx - t * Dd;

  const float hsv = hs_conv[idx];
  const float dtv = dt[idx];
  const float gate = proj[(size_t)t * PROJN + Dd + d];
  const float hd = hsv * dtv;

  const v4f* ssin = (const v4f*)(ssm_in + (size_t)idx * Ss);
  v4f* ssout = (v4f*)(ss_out + (size_t)idx * Ss);
  const v4f* al = (const v4f*)(A_log + (size_t)d * Ss);
  const v4f* Bm = (const v4f*)(sp + (size_t)t * SPN + Rr);
  const v4f* Cm = (const v4f*)(sp + (size_t)t * SPN + Rr + Ss);

  float y = 0.f;
#pragma unroll
  for (int j = 0; j < 4; ++j) {
    v4f a = al[j];
    v4f b = Bm[j];
    v4f c = Cm[j];
    v4f s0 = __builtin_nontemporal_load(ssin + j);
    v4f ns;
    ns.x = s0.x * __expf(dtv * (-__expf(a.x))) + hd * b.x;
    ns.y = s0.y * __expf(dtv * (-__expf(a.y))) + hd * b.y;
    ns.z = s0.z * __expf(dtv * (-__expf(a.z))) + hd * b.z;
    ns.w = s0.w * __expf(dtv * (-__expf(a.w))) + hd * b.w;
    y += ns.x * c.x + ns.y * c.y + ns.z * c.z + ns.w * c.w;
    __builtin_nontemporal_store(ns, ssout + j);
  }
  float yv = y + D_param[d] * hsv;
  yg[idx] = yv * silu_f(gate);
}

// ---------------------------------------------------------------------------
extern "C" void kernel_launch(void* const* d_in, const int* in_sizes, int n_in,
                              void* d_out, int out_size, void* d_ws, size_t ws_size,
                              hipStream_t stream) {
  const float* u           = (const float*)d_in[0];
  // d_in[1] = h (unused by reference)
  const float* conv_states = (const float*)d_in[2];
  const float* ssm_states  = (const float*)d_in[3];
  const float* w_in        = (const float*)d_in[4];
  const float* conv_w      = (const float*)d_in[5];
  const float* conv_b      = (const float*)d_in[6];
  const float* w_x         = (const float*)d_in[7];
  const float* w_dt        = (const float*)d_in[8];
  const float* b_dt        = (const float*)d_in[9];
  const float* A_log       = (const float*)d_in[10];
  const float* D_param     = (const float*)d_in[11];
  const float* w_out       = (const float*)d_in[12];
  const float* norm_w      = (const float*)d_in[13];
  const float* normf_w     = (const float*)d_in[14];

  // Workspace layout (~36 MB of fp32)
  float* ws  = (float*)d_ws;
  float* hid = ws;  ws += (size_t)Tt * Hh;
  float* xn  = ws;  ws += (size_t)Tt * Hh;
  float* proj = ws; ws += (size_t)Tt * PROJN;
  float* hs  = ws;  ws += (size_t)Tt * Dd;
  float* sp  = ws;  ws += (size_t)Tt * SPN;
  float* dt  = ws;  ws += (size_t)Tt * Dd;
  float* yg  = ws;  ws += (size_t)Tt * Dd;

  // Output layout: [out (T*H)] [new_cs (L*T*D*K)] [new_ss (L*T*D*S)]
  float* out0   = (float*)d_out;
  float* cs_out = out0 + (size_t)Tt * Hh;
  float* ss_out = cs_out + (size_t)Ll * Tt * Dd * Kk;

  // hid = u (flattened (T,H))
  (void)hipMemcpyAsync(hid, u, (size_t)Tt * Hh * sizeof(float),
                       hipMemcpyDeviceToDevice, stream);

  const int elemBlocks = (Tt * Dd) / 256;

  for (int l = 0; l < Ll; ++l) {
    // xn = rms(hid) * norm_w[l]
    rmsnorm_kernel<<<Tt, 256, 0, stream>>>(hid, norm_w + (size_t)l * Hh, xn, Hh);

    // proj = xn @ w_in[l]^T   (1024 x 768 x 3072), 16x64 strip per wave
    wmma_gemm_nt<0, 4><<<(Tt / 16) * (PROJN / 64) / 8, 256, 0, stream>>>(
        xn, w_in + (size_t)l * PROJN * Hh, proj, nullptr, Tt, PROJN, Hh, Hh);

    // depthwise conv + silu; emit new conv state (NT streams)
    conv_kernel<<<elemBlocks, 256, 0, stream>>>(
        proj, conv_states + (size_t)l * Tt * Dd * Kk,
        conv_w + (size_t)l * Dd * Kk, conv_b + (size_t)l * Dd, hs,
        cs_out + (size_t)l * Tt * Dd * Kk);

    // sp = hs @ w_x[l]^T   (1024 x 1536 x 80): N=80 -> single-tile waves
    wmma_gemm_nt<0, 1><<<(Tt / 16) * (SPN / 16) / 8, 256, 0, stream>>>(
        hs, w_x + (size_t)l * SPN * Dd, sp, nullptr, Tt, SPN, Dd, Dd);

    // dt = softplus(sp[:, :R] @ w_dt[l]^T + b_dt[l])  (1024 x 48 x 1536)
    wmma_gemm_nt<1, 4><<<(Tt / 16) * (Dd / 64) / 8, 256, 0, stream>>>(
        sp, w_dt + (size_t)l * Dd * Rr, dt, b_dt + (size_t)l * Dd,
        Tt, Dd, Rr, SPN);

    // SSM state update + gated y (BW-critical, NT 128b streams)
    ssm_kernel<<<elemBlocks, 256, 0, stream>>>(
        hs, dt, sp, proj, ssm_states + (size_t)l * Tt * Dd * Ss,
        A_log + (size_t)l * Dd * Ss, D_param + (size_t)l * Dd, yg,
        ss_out + (size_t)l * Tt * Dd * Ss);

    // hid += yg @ w_out[l]^T   (1024 x 1536 x 768), residual fused
    wmma_gemm_nt<2, 4><<<(Tt / 16) * (Hh / 64) / 8, 256, 0, stream>>>(
        yg, w_out + (size_t)l * Hh * Dd, hid, nullptr, Tt, Hh, Dd, Dd);
  }

  // out = rms(hid) * normf_w
  rmsnorm_kernel<<<Tt, 256, 0, stream>>>(hid, normf_w, out0, Hh);
}